// tPTPriorNetwork_42133629174458
// MI455X (gfx1250) — compile-verified
//
#include <hip/hip_runtime.h>

typedef __attribute__((ext_vector_type(2))) float v2f;
typedef __attribute__((ext_vector_type(8))) float v8f;

#define BATCH 256
#define DIM   512
#define NCB   100000
#define NPAIR (NCB/32)   // 3125 pairs of 16-wide column tiles

// ---------------------------------------------------------------------------
// VALU butterfly min over each 16-lane half (v_permlane16_b32, no LDS).
// ---------------------------------------------------------------------------
__device__ __forceinline__ unsigned halfmin16(unsigned v) {
  unsigned t;
  t = __builtin_amdgcn_permlane16(v, v, 0x67452301u, 0xEFCDAB89u, false, false); // ^1
  v = t < v ? t : v;
  t = __builtin_amdgcn_permlane16(v, v, 0x54761032u, 0xDCFE98BAu, false, false); // ^2
  v = t < v ? t : v;
  t = __builtin_amdgcn_permlane16(v, v, 0x32107654u, 0xBA98FEDCu, false, false); // ^4
  v = t < v ? t : v;
  t = __builtin_amdgcn_permlane16(v, v, 0xFEDCBA98u, 0x76543210u, false, false); // ^8
  v = t < v ? t : v;
  return v;
}

// cross-half (lane XOR 16) exchange via v_permlanex16_b32 (identity selector)
__device__ __forceinline__ float xhalf_add(float x) {
  unsigned u = __float_as_uint(x);
  unsigned o = __builtin_amdgcn_permlanex16(u, u, 0x76543210u, 0xFEDCBA98u, false, false);
  return x + __uint_as_float(o);
}

// argmin epilogue for one 16x(16x16-tiles-stack) accumulator set
__device__ __forceinline__ void reduce_tile(
    const v8f* acc, float bnorm, int ncolbase,
    int mhalf, int ln, int lh, unsigned long long* best)
{
#pragma unroll
  for (int mt = 0; mt < 8; ++mt) {
#pragma unroll
    for (int v = 0; v < 8; ++v) {
      float score = bnorm - 2.0f * acc[mt][v];
      unsigned sb = __float_as_uint(score);
      sb ^= (sb >> 31) ? 0xFFFFFFFFu : 0x80000000u;   // order-preserving map
      const unsigned mn = halfmin16(sb);              // min over this half's lanes
      const unsigned long long bal = __ballot(sb == mn);
      const unsigned mask = (unsigned)(bal >> (lh * 16)) & 0xFFFFu;
      if (ln == 0) {                                  // lanes 0 (M=v), 16 (M=v+8)
        const int nl = __ffs(mask) - 1;               // lowest index on ties
        const unsigned long long pack =
            ((unsigned long long)mn << 32) | (unsigned)(ncolbase + nl);
        const int m = (mhalf * 8 + mt) * 16 + v + lh * 8;
        atomicMin(&best[m], pack);
      }
    }
  }
}

// ---------------------------------------------------------------------------
// Kernel 0: init packed (score,index) argmin results
// ---------------------------------------------------------------------------
__global__ __launch_bounds__(256) void init_result(unsigned long long* __restrict__ r) {
  r[threadIdx.x] = ~0ull;
}

// ---------------------------------------------------------------------------
// Kernel 1: fused distance GEMM + argmin over N.
// score(b,n) = ||cb_n||^2 - 2 * codes_b . cb_n
// Each wave: 8 row-tiles (128 code rows) x 2 adjacent 16-col codebook tiles.
// A-fragments are shared by both B tiles: 10 loads per 16 wmma.
// ---------------------------------------------------------------------------
__global__ __launch_bounds__(256) void knn_argmin(
    const float* __restrict__ codes,         // [256,512]
    const float* __restrict__ cb,            // [100000,512]
    unsigned long long* __restrict__ result) // [256] (orderedF32<<32)|idx
{
  __shared__ unsigned long long best[BATCH];
  const int tid = threadIdx.x;
  best[tid] = ~0ull;
  __syncthreads();

  const int wave  = tid >> 5;
  const int lane  = tid & 31;
  const int mhalf = wave & 1;   // which 128 code rows
  const int nidx  = wave >> 1;  // 0..3: column-pair slot within block
  const int ln    = lane & 15;
  const int lh    = lane >> 4;

  const float* xbase = codes + (size_t)(mhalf * 128 + ln) * DIM + lh * 2;

  for (int p = blockIdx.x * 4 + nidx; p < NPAIR; p += gridDim.x * 4) {
    const int n0 = p * 32;
    const float* bb0 = cb + (size_t)(n0 + ln) * DIM + lh * 2;
    const float* bb1 = cb + (size_t)(n0 + 16 + ln) * DIM + lh * 2;

    v8f acc0[8] = {}, acc1[8] = {};
    float bn0 = 0.0f, bn1 = 0.0f;
#pragma unroll 2
    for (int k = 0; k < DIM; k += 4) {
      v2f bf0 = __builtin_nontemporal_load((const v2f*)(bb0 + k));  // streamed once
      v2f bf1 = __builtin_nontemporal_load((const v2f*)(bb1 + k));
      bn0 += bf0.x * bf0.x + bf0.y * bf0.y;
      bn1 += bf1.x * bf1.x + bf1.y * bf1.y;
#pragma unroll
      for (int mt = 0; mt < 8; ++mt) {
        v2f af = *(const v2f*)(xbase + (size_t)mt * 16 * DIM + k);
        acc0[mt] = __builtin_amdgcn_wmma_f32_16x16x4_f32(
            false, af, false, bf0, (short)0, acc0[mt], false, false);
        acc1[mt] = __builtin_amdgcn_wmma_f32_16x16x4_f32(
            false, af, false, bf1, (short)0, acc1[mt], false, false);
      }
    }
    reduce_tile(acc0, xhalf_add(bn0), n0,      mhalf, ln, lh, best);
    reduce_tile(acc1, xhalf_add(bn1), n0 + 16, mhalf, ln, lh, best);
  }
  __syncthreads();
  if (best[tid] != ~0ull) atomicMin(&result[tid], best[tid]);
}

// ---------------------------------------------------------------------------
// Kernel 2: prev[b] = codebook[argmin_idx[b]]
// ---------------------------------------------------------------------------
__global__ __launch_bounds__(128) void gather_rows(
    const unsigned long long* __restrict__ result,
    const float* __restrict__ cb, float* __restrict__ prev)
{
  const int b = blockIdx.x;
  const unsigned idx = (unsigned)(result[b] & 0xFFFFFFFFull);
  const float4* src = (const float4*)(cb + (size_t)idx * DIM);
  float4* dst = (float4*)(prev + (size_t)b * DIM);
  dst[threadIdx.x] = src[threadIdx.x];
}

// ---------------------------------------------------------------------------
// Kernel 3: generic layer  Y = act( (X [+X2]) @ W^T + bias )  [+= existing Y]
// ---------------------------------------------------------------------------
template <bool TANH, bool OUTADD>
__global__ __launch_bounds__(128) void gemm_tile(
    const float* __restrict__ X, const float* __restrict__ X2,
    const float* __restrict__ W, const float* __restrict__ bias,
    float* __restrict__ Y)
{
  const int tid  = threadIdx.x;
  const int wave = tid >> 5, lane = tid & 31;
  const int ln = lane & 15, lh = lane >> 4;
  const int tile = blockIdx.x * 4 + wave;  // 0..511
  const int mt = tile & 15;                // row tile (batch)
  const int ot = tile >> 4;                // col tile (out features)

  const float* wp  = W + (size_t)(ot * 16 + ln) * DIM + lh * 2;  // W[o][k] = (W^T)[k][o]
  const float* xp  = X + (size_t)(mt * 16 + ln) * DIM + lh * 2;
  const float* x2p = X2 ? X2 + (size_t)(mt * 16 + ln) * DIM + lh * 2 : nullptr;

  v8f acc[4] = {};
  for (int k = 0; k < DIM; k += 16) {
#pragma unroll
    for (int j = 0; j < 4; ++j) {
      v2f bf = *(const v2f*)(wp + k + 4 * j);
      v2f af = *(const v2f*)(xp + k + 4 * j);
      if (X2) {
        v2f a2 = *(const v2f*)(x2p + k + 4 * j);
        af.x += a2.x; af.y += a2.y;
      }
      acc[j] = __builtin_amdgcn_wmma_f32_16x16x4_f32(
          false, af, false, bf, (short)0, acc[j], false, false);
    }
  }
  v8f sum = (acc[0] + acc[1]) + (acc[2] + acc[3]);

  const int ocol = ot * 16 + ln;
  const float bv = bias[ocol];
#pragma unroll
  for (int v = 0; v < 8; ++v) {
    const int m = mt * 16 + v + lh * 8;      // C layout: M = v + 8*(lane/16)
    float y = sum[v] + bv;
    if (TANH) y = tanhf(y);
    float* p = Y + (size_t)m * DIM + ocol;
    if (OUTADD) y += *p;
    *p = y;
  }
}

// ---------------------------------------------------------------------------
extern "C" void kernel_launch(void* const* d_in, const int* in_sizes, int n_in,
                              void* d_out, int out_size, void* d_ws, size_t ws_size,
                              hipStream_t stream)
{
  (void)in_sizes; (void)n_in; (void)out_size; (void)ws_size;
  const float* codes = (const float*)d_in[0];
  const float* cb    = (const float*)d_in[1];
  const float* W_in  = (const float*)d_in[2];  const float* b_in  = (const float*)d_in[3];
  const float* W_h1  = (const float*)d_in[4];  const float* b_h1  = (const float*)d_in[5];
  const float* W_s2  = (const float*)d_in[6];  const float* b_s2  = (const float*)d_in[7];
  const float* W_s3  = (const float*)d_in[8];  const float* b_s3  = (const float*)d_in[9];
  const float* W_so1 = (const float*)d_in[10]; const float* b_so1 = (const float*)d_in[11];
  const float* W_so2 = (const float*)d_in[12]; const float* b_so2 = (const float*)d_in[13];
  const float* W_h2  = (const float*)d_in[14]; const float* b_h2  = (const float*)d_in[15];
  const float* W_h3  = (const float*)d_in[16]; const float* b_h3  = (const float*)d_in[17];
  const float* W_mu  = (const float*)d_in[18]; const float* b_mu  = (const float*)d_in[19];
  const float* W_ls  = (const float*)d_in[20]; const float* b_ls  = (const float*)d_in[21];

  const size_t MAT = (size_t)BATCH * DIM;  // 131072 floats
  unsigned long long* result = (unsigned long long*)d_ws;
  float* prev = (float*)((char*)d_ws + 4096);
  float* t_i  = prev + MAT;
  float* t_h1 = t_i  + MAT;
  float* t_s2 = t_h1 + MAT;
  float* t_s3 = t_s2 + MAT;
  float* t_h2 = t_s3 + MAT;
  float* t_os = t_h2 + MAT;
  float* mu = (float*)d_out;
  float* ls = mu + MAT;

  init_result<<<1, 256, 0, stream>>>(result);
  knn_argmin <<<391, 256, 0, stream>>>(codes, cb, result);  // 391*4 slots -> 2 passes
  gather_rows<<<BATCH, 128, 0, stream>>>(result, cb, prev);

  gemm_tile<true,  false><<<128, 128, 0, stream>>>(prev, nullptr, W_in,  b_in,  t_i);
  gemm_tile<true,  false><<<128, 128, 0, stream>>>(t_i,  nullptr, W_h1,  b_h1,  t_h1);
  gemm_tile<true,  false><<<128, 128, 0, stream>>>(t_h1, nullptr, W_s2,  b_s2,  t_s2);
  gemm_tile<true,  false><<<128, 128, 0, stream>>>(t_h1, nullptr, W_s3,  b_s3,  t_s3);
  gemm_tile<true,  false><<<128, 128, 0, stream>>>(t_h1, t_s2,    W_h2,  b_h2,  t_h2);
  gemm_tile<true,  false><<<128, 128, 0, stream>>>(t_h1, nullptr, W_so1, b_so1, t_os);
  gemm_tile<true,  true ><<<128, 128, 0, stream>>>(t_h2, nullptr, W_so2, b_so2, t_os);
  gemm_tile<true,  true ><<<128, 128, 0, stream>>>(t_h2, t_s3,    W_h3,  b_h3,  t_os);
  gemm_tile<false, false><<<128, 128, 0, stream>>>(t_os, nullptr, W_mu,  b_mu,  mu);
  gemm_tile<false, false><<<128, 128, 0, stream>>>(t_os, nullptr, W_ls,  b_ls,  ls);
}